// RoiAlign_61701500174898
// MI455X (gfx1250) — compile-verified
//
#include <hip/hip_runtime.h>
#include <hip/hip_bf16.h>

// ROI Align (tf.image.crop_and_resize, bilinear, extrapolation_value=0)
// fm: [8,64,64,256] f32, boxes: [8,1000,4] f32 (y1,x1,y2,x2), out: [8,1000,7,7,256] f32
//
// Separable bilinear as WMMA: per box, per PAIR of output rows (py0,py1):
//   out[(py,px), c] = sum_{k=0..31} W[(py,px), k] * Gy[k, c]
// Gy k in [0,16): y-lerped gather columns for py0; k in [16,32): for py1.
// W rows 0-6 -> py0 (nonzero at k=2px+{0,1}), rows 8-14 -> py1 (k=16+2px+{0,1}).
// Validity masks are folded into the wx / wy weights (reproduces where(valid,.,0)).
// Eight K-steps of V_WMMA_F32_16X16X4_F32 per 16-channel tile; both half-waves
// of the D matrix map to live output rows (lanes 0-15 -> py0, 16-31 -> py1).

typedef __attribute__((ext_vector_type(2))) float v2f;
typedef __attribute__((ext_vector_type(8))) float v8f;

#define HDIM 64
#define WDIM 64
#define CDIM 256
#define PP 7
#define NBOX 1000
#define GSTRIDE 272   // 256 + 16 pad: +16 banks per row -> half-wave B reads conflict-free

__global__ __launch_bounds__(128) void roialign_wmma_kernel(
    const float* __restrict__ fm, const float* __restrict__ boxes,
    float* __restrict__ out)
{
    __shared__ float sG[32 * GSTRIDE];          // Gy[k][c]: 2 py rows x 16 cols, y-lerped
    __shared__ float sWx0[16], sWx1[16];        // (1-wx)*vx, wx*vx  (0 for px>=7)
    __shared__ int   sXlo[16], sXhi[16];
    __shared__ int   sYlo[8],  sYhi[8];
    __shared__ float sWy0[8],  sWy1[8];         // (1-wy)*vy, wy*vy  (0 for py>=7)

    const int tid   = threadIdx.x;
    const int lane  = tid & 31;
    const int wave  = tid >> 5;
    const int m     = lane & 15;                // M row (px slot) / N col (channel)
    const int hi    = (lane >> 4) & 1;          // half-wave: 0 -> M 0-7, 1 -> M 8-15
    const int sel   = m >> 3;                   // which py block this M row belongs to
    const int local = m & 7;                    // px within the block (7 == padding)

    const int bn = blockIdx.x;                  // b*1000 + n
    const int b  = bn / NBOX;
    const float* bx = boxes + (size_t)bn * 4;

    if (tid < 16) { sWx0[tid] = 0.f; sWx1[tid] = 0.f; sXlo[tid] = 0; sXhi[tid] = 0; }
    if (tid < 8)  { sYlo[tid] = 0;   sYhi[tid] = 0;   sWy0[tid] = 0.f; sWy1[tid] = 0.f; }
    if (tid < PP) {
        const float y1 = bx[0], x1 = bx[1], y2 = bx[2], x2 = bx[3];
        const float t  = (float)tid;
        const float yc = y1 * 63.f + t * ((y2 - y1) * 63.f / 6.f);
        const float xc = x1 * 63.f + t * ((x2 - x1) * 63.f / 6.f);
        const float yf = floorf(yc), xf = floorf(xc);
        const float fy = yc - yf,    fx = xc - xf;
        sYlo[tid] = min(max((int)yf, 0), HDIM - 1);
        sYhi[tid] = min(max((int)yf + 1, 0), HDIM - 1);
        sXlo[tid] = min(max((int)xf, 0), WDIM - 1);
        sXhi[tid] = min(max((int)xf + 1, 0), WDIM - 1);
        const bool vy = (yc >= 0.f) && (yc <= (float)(HDIM - 1));
        const bool vx = (xc >= 0.f) && (xc <= (float)(WDIM - 1));
        sWy0[tid] = vy ? (1.f - fy) : 0.f;
        sWy1[tid] = vy ? fy : 0.f;
        sWx0[tid] = vx ? (1.f - fx) : 0.f;
        sWx1[tid] = vx ? fx : 0.f;
    }
    __syncthreads();

    // A fragments (constant across the whole box): W[m, k], 8 K-chunks of 4.
    // ISA 16x4 f32 A layout: lanes 0-15 hold K={k0,k0+1} in VGPR{0,1}; lanes 16-31 K={k0+2,k0+3}.
    v2f A[8];
#pragma unroll
    for (int j = 0; j < 8; ++j) {
        const int k0 = 4 * j + 2 * hi;
        const int k1 = k0 + 1;
        const bool h0 = ((k0 >> 4) == sel) && (((k0 >> 1) & 7) == local);
        const bool h1 = ((k1 >> 4) == sel) && (((k1 >> 1) & 7) == local);
        A[j].x = h0 ? ((k0 & 1) ? sWx1[local] : sWx0[local]) : 0.f;
        A[j].y = h1 ? ((k1 & 1) ? sWx1[local] : sWx0[local]) : 0.f;
    }

    const float* fmb  = fm + (size_t)b * (HDIM * WDIM * CDIM);
    float*       outb = out + (size_t)bn * (PP * PP * CDIM);

    for (int p = 0; p < 4; ++p) {
        const int py0 = 2 * p, py1 = 2 * p + 1;          // py1 == 7 -> zero weights
        const int   yl0 = sYlo[py0], yh0 = sYhi[py0];
        const int   yl1 = sYlo[py1], yh1 = sYhi[py1];
        const float a00 = sWy0[py0], a01 = sWy1[py0];
        const float a10 = sWy0[py1], a11 = sWy1[py1];

        // Stage Gy into LDS: 32 rows x 256 ch = 2048 float4 tasks / 128 threads.
#pragma unroll
        for (int it = 0; it < 16; ++it) {
            const int idx = it * 128 + tid;
            const int k   = idx >> 6;                    // 0..31
            const int c   = (idx & 63) * 4;
            const int kk  = k & 15;
            const int ps  = k >> 4;                      // 0 -> py0, 1 -> py1
            const int px  = kk >> 1;
            const int xc  = (kk < 14) ? ((kk & 1) ? sXhi[px] : sXlo[px]) : 0;
            const int   yl = ps ? yl1 : yl0;
            const int   yh = ps ? yh1 : yh0;
            const float a0 = ps ? a10 : a00;
            const float a1 = ps ? a11 : a01;
            const float4 lo = *(const float4*)(fmb + ((size_t)yl * WDIM + xc) * CDIM + c);
            const float4 hv = *(const float4*)(fmb + ((size_t)yh * WDIM + xc) * CDIM + c);
            float4 g;
            g.x = a0 * lo.x + a1 * hv.x;
            g.y = a0 * lo.y + a1 * hv.y;
            g.z = a0 * lo.z + a1 * hv.z;
            g.w = a0 * lo.w + a1 * hv.w;
            *(float4*)(&sG[k * GSTRIDE + c]) = g;
        }
        __syncthreads();

        // Prefetch the next pair's four feature rows while the matrix pipe works.
        if (p + 1 < 4) {
            const int k   = tid >> 3;                    // 16 columns
            const int seg = tid & 7;                     // 8 x 128B per 1KB column
            const int px  = k >> 1;
            const int xc  = (k < 14) ? ((k & 1) ? sXhi[px] : sXlo[px]) : 0;
            const size_t coff = (size_t)xc * CDIM + (size_t)seg * 32;
            __builtin_prefetch(fmb + (size_t)sYlo[py0 + 2] * (WDIM * CDIM) + coff, 0, 3);
            __builtin_prefetch(fmb + (size_t)sYhi[py0 + 2] * (WDIM * CDIM) + coff, 0, 3);
            __builtin_prefetch(fmb + (size_t)sYlo[py1 + 2] * (WDIM * CDIM) + coff, 0, 3);
            __builtin_prefetch(fmb + (size_t)sYhi[py1 + 2] * (WDIM * CDIM) + coff, 0, 3);
        }

        // WMMA: each wave owns 4 channel tiles; 8 K-steps per tile.
        // B 4x16 f32 layout (mirror of A): lanes 0-15 hold K={k0,k0+1}, lanes 16-31 K={k0+2,k0+3}.
#pragma unroll
        for (int t4 = 0; t4 < 4; ++t4) {
            const int col = (wave * 4 + t4) * 16 + m;
            v8f acc = {0.f, 0.f, 0.f, 0.f, 0.f, 0.f, 0.f, 0.f};
#pragma unroll
            for (int j = 0; j < 8; ++j) {
                const int k0 = 4 * j + 2 * hi;
                v2f bv;
                bv.x = sG[k0 * GSTRIDE + col];
                bv.y = sG[(k0 + 1) * GSTRIDE + col];
                acc = __builtin_amdgcn_wmma_f32_16x16x4_f32(
                    /*neg_a=*/false, A[j], /*neg_b=*/false, bv,
                    /*c_mod=*/(short)0, acc, /*reuse_a=*/false, /*reuse_b=*/false);
            }
            // D layout: VGPR r -> M=r (lanes 0-15, py0 row) / M=8+r (lanes 16-31, py1 row).
            const int mypy = hi ? py1 : py0;
            float* obase = outb + (size_t)mypy * (PP * CDIM);
            if (mypy < PP) {                              // false only for py==7 half-wave
#pragma unroll
                for (int r = 0; r < PP; ++r)
                    __builtin_nontemporal_store(acc[r], obase + (size_t)r * CDIM + col);
            }
        }
        __syncthreads();
    }
}

extern "C" void kernel_launch(void* const* d_in, const int* in_sizes, int n_in,
                              void* d_out, int out_size, void* d_ws, size_t ws_size,
                              hipStream_t stream) {
    (void)in_sizes; (void)n_in; (void)d_ws; (void)ws_size; (void)out_size;
    const float* fm    = (const float*)d_in[0];   // [8,64,64,256] f32
    const float* boxes = (const float*)d_in[1];   // [8,1000,4]    f32
    // d_in[2] = pool_size (==7), compile-time specialized
    float* out = (float*)d_out;                   // [8,1000,7,7,256] f32

    dim3 grid(8 * NBOX);
    dim3 block(128);
    roialign_wmma_kernel<<<grid, block, 0, stream>>>(fm, boxes, out);
}